// CorrespondenceContrastiveLoss_86517821210742
// MI455X (gfx1250) — compile-verified
//
#include <hip/hip_runtime.h>

// CDNA5 / gfx1250: wave32, WMMA (not MFMA).
typedef __attribute__((ext_vector_type(2))) float v2f;
typedef __attribute__((ext_vector_type(8))) float v8f;

#define CX 100
#define CY 88
#define CZ 80
#define NCH 64
#define BATCH 4096
#define CH_STRIDE ((size_t)CX * CY * CZ)   /* 704000 elements between channels */
#define SPATIAL_YZ (CY * CZ)               /* 7040 */

__device__ __forceinline__ int wrap_mod(int v, int m) {
    // points are in [0, 128); m in {100, 88, 80} -> at most one subtraction
    return (v >= m) ? (v - m) : v;
}

// One wave (32 lanes) per tile of 16 batch elements.
// d2_pos/d2_neg computed as the diagonal of diff * diff^T via fp32 WMMA
// (V_WMMA_F32_16X16X4_F32), accumulating 16 chunks of K=4 channels.
__global__ void __launch_bounds__(32)
ccl_main_kernel(const float* __restrict__ fix,
                const float* __restrict__ mov,
                const int* __restrict__ fpts,
                const int* __restrict__ ppts,
                const int* __restrict__ npts,
                float* __restrict__ out,       // [1 + 2*BATCH]
                float* __restrict__ partial)   // [gridDim.x]
{
    const int lane = threadIdx.x;   // 0..31
    const int r    = lane & 15;     // row of the 16-row tile this lane feeds
    const int h    = lane >> 4;     // K-half selector (A/B f32 layout)
    const int tile = blockIdx.x;    // 0..255
    const int b    = tile * 16 + r;

    // point_redirection: coord % crop
    const int fx = wrap_mod(fpts[b * 3 + 0], CX);
    const int fy = wrap_mod(fpts[b * 3 + 1], CY);
    const int fz = wrap_mod(fpts[b * 3 + 2], CZ);
    const int px = wrap_mod(ppts[b * 3 + 0], CX);
    const int py = wrap_mod(ppts[b * 3 + 1], CY);
    const int pz = wrap_mod(ppts[b * 3 + 2], CZ);
    const int nx = wrap_mod(npts[b * 3 + 0], CX);
    const int ny = wrap_mod(npts[b * 3 + 1], CY);
    const int nz = wrap_mod(npts[b * 3 + 2], CZ);

    const size_t sf = (size_t)fx * SPATIAL_YZ + (size_t)fy * CZ + (size_t)fz;
    const size_t sp = (size_t)px * SPATIAL_YZ + (size_t)py * CZ + (size_t)pz;
    const size_t sn = (size_t)nx * SPATIAL_YZ + (size_t)ny * CZ + (size_t)nz;

    v8f cP = {};  // Gram accumulator for (f - p)
    v8f cN = {};  // Gram accumulator for (f - n)

    // Per chunk j: lane supplies diff[r][c0], diff[r][c0+1] with c0 = 4j + 2h.
    // Both A (16x4) and B (4x16) f32 fragments index the 16-dim by lane&15 and
    // K by {lane>>4, vgpr} identically, so the same v2f serves as A and B=A^T.
    #pragma unroll
    for (int j = 0; j < 16; ++j) {
        const int    c0 = 4 * j + 2 * h;
        const size_t o0 = (size_t)c0 * CH_STRIDE;
        const size_t o1 = o0 + CH_STRIDE;

        const float f0 = fix[o0 + sf];
        const float f1 = fix[o1 + sf];
        const float p0 = mov[o0 + sp];
        const float p1 = mov[o1 + sp];
        const float n0 = mov[o0 + sn];
        const float n1 = mov[o1 + sn];

        v2f dP; dP.x = f0 - p0; dP.y = f1 - p1;
        v2f dN; dN.x = f0 - n0; dN.y = f1 - n1;

        // 8 args: (neg_a, A, neg_b, B, c_mod, C, reuse_a, reuse_b)
        cP = __builtin_amdgcn_wmma_f32_16x16x4_f32(
                 false, dP, false, dP, (short)0, cP, false, false);
        cN = __builtin_amdgcn_wmma_f32_16x16x4_f32(
                 false, dN, false, dN, (short)0, cN, false, false);
    }

    // Diagonal D[m][m]: m<8 -> (lane m, vgpr m); m>=8 -> (lane m+16, vgpr m-8).
    const bool active = (lane < 8) || (lane >= 24);
    const int  vi     = (lane < 8) ? lane : (lane - 24);  // vgpr index (active lanes)

    float d2p = 0.0f, d2n = 0.0f;
    #pragma unroll
    for (int v = 0; v < 8; ++v) {
        if (vi == v) { d2p = cP[v]; d2n = cN[v]; }
    }

    float contrib = 0.0f;
    if (active) {
        // For active lanes the diagonal row m equals this lane's own r (and b).
        const float pd = sqrtf(d2p);
        const float nd = sqrtf(d2n);
        out[1 + b]         = pd;   // pos_dis
        out[1 + BATCH + b] = nd;   // neg_dis
        const float hg = fmaxf(0.0f, 1.0f - nd);   // MARGIN = 1.0
        contrib = d2p + hg * hg;
    }

    // wave32 butterfly reduction of the loss contribution
    #pragma unroll
    for (int off = 16; off > 0; off >>= 1)
        contrib += __shfl_xor(contrib, off, 32);

    if (lane == 0)
        partial[tile] = contrib;
}

// Deterministic tree reduction of per-tile partials -> scalar loss.
__global__ void __launch_bounds__(256)
ccl_reduce_kernel(const float* __restrict__ partial,
                  float* __restrict__ out, int n)
{
    __shared__ float s[256];
    const int t = threadIdx.x;
    s[t] = (t < n) ? partial[t] : 0.0f;
    __syncthreads();
    #pragma unroll
    for (int off = 128; off > 0; off >>= 1) {
        if (t < off) s[t] += s[t + off];
        __syncthreads();
    }
    if (t == 0)
        out[0] = s[0] * (100.0f / (4.0f * (float)BATCH));  // /(4B) * 100
}

extern "C" void kernel_launch(void* const* d_in, const int* in_sizes, int n_in,
                              void* d_out, int out_size, void* d_ws, size_t ws_size,
                              hipStream_t stream) {
    (void)in_sizes; (void)n_in; (void)out_size; (void)ws_size;

    const float* fix  = (const float*)d_in[0];  // [1,64,100,88,80] f32
    const float* mov  = (const float*)d_in[1];  // [1,64,100,88,80] f32
    const int*   fpts = (const int*)d_in[2];    // [4096,3] i32
    const int*   ppts = (const int*)d_in[3];    // [4096,3] i32
    const int*   npts = (const int*)d_in[4];    // [4096,3] i32
    // d_in[5..7] = x/y/z_shard: unused by the reference math.

    float* out     = (float*)d_out;             // [1 + 4096 + 4096]
    float* partial = (float*)d_ws;              // 256 floats of scratch

    const int tiles = BATCH / 16;               // 256 waves, spread across WGPs
    ccl_main_kernel<<<tiles, 32, 0, stream>>>(fix, mov, fpts, ppts, npts, out, partial);
    ccl_reduce_kernel<<<1, 256, 0, stream>>>(partial, out, tiles);
}